// SparseGrid_57526791963271
// MI455X (gfx1250) — compile-verified
//
#include <hip/hip_runtime.h>
#include <stdint.h>
#include <stddef.h>

#define RES       192
#define SH_DIM    27
#define BLOCK     256
#define WAVE      32
#define WPB       (BLOCK / WAVE)         // 8 waves per block
#define DW_PER_WV (WAVE * SH_DIM)        // 864 dwords staged per wave
#define V4_PER_WV (DW_PER_WV / 4)        // 216 float4 chunks per wave

typedef float f4 __attribute__((ext_vector_type(4)));
// sh rows start at idx*108 B (only dword aligned). gfx1250 supports unaligned
// b128; packed struct tells clang align==1 so it still emits wide loads.
struct __attribute__((packed)) f4u { f4 v; };

// Unconditional gather + weighted accumulate (branchless: w==0 for invalid
// links, matching the reference's where(valid, data, 0) semantics).
__device__ __forceinline__ void gather_acc(float w, uint32_t idx,
                                           const float* __restrict__ density,
                                           const float* __restrict__ sh,
                                           float& sigma, float* acc) {
    sigma = fmaf(w, density[idx], sigma);
    const float* row = sh + idx * (uint32_t)SH_DIM;   // 32-bit offset -> saddr+voffset form
    const f4u* vp = (const f4u*)row;
#pragma unroll
    for (int j = 0; j < 6; ++j) {
        f4 v = vp[j].v;
        acc[j * 4 + 0] = fmaf(w, v.x, acc[j * 4 + 0]);
        acc[j * 4 + 1] = fmaf(w, v.y, acc[j * 4 + 1]);
        acc[j * 4 + 2] = fmaf(w, v.z, acc[j * 4 + 2]);
        acc[j * 4 + 3] = fmaf(w, v.w, acc[j * 4 + 3]);
    }
    acc[24] = fmaf(w, row[24], acc[24]);
    acc[25] = fmaf(w, row[25], acc[25]);
    acc[26] = fmaf(w, row[26], acc[26]);
}

__global__ __launch_bounds__(BLOCK) void trilerp_sh_kernel(
    const float* __restrict__ points,   // (N,3)
    const float* __restrict__ density,  // (CAP,1)
    const float* __restrict__ sh,       // (CAP,27)
    const int*   __restrict__ links,    // (192,192,192)
    float* __restrict__ out_sigma,      // (N)
    float* __restrict__ out_rgb,        // (N,27)
    int npts) {
    // Per-wave rgb staging for coalesced async writeout. Row = 3456 B (16B aligned).
    __shared__ __align__(16) float stage[WPB][DW_PER_WV];

    const int tid  = threadIdx.x;
    const int lane = tid & (WAVE - 1);
    const int wv   = tid >> 5;
    const int pt   = blockIdx.x * BLOCK + tid;
    const bool active = pt < npts;

    float sigma = 0.0f;
    float acc[SH_DIM];
#pragma unroll
    for (int c = 0; c < SH_DIM; ++c) acc[c] = 0.0f;

    if (active) {
        const float px = points[(size_t)pt * 3 + 0];
        const float py = points[(size_t)pt * 3 + 1];
        const float pz = points[(size_t)pt * 3 + 2];
        // p = 0.5*gsz - 0.5 + x * 0.5*gsz, clamped to [0, gsz-1]
        float fx = fminf(fmaxf(fmaf(px, 96.0f, 95.5f), 0.0f), 191.0f);
        float fy = fminf(fmaxf(fmaf(py, 96.0f, 95.5f), 0.0f), 191.0f);
        float fz = fminf(fmaxf(fmaf(pz, 96.0f, 95.5f), 0.0f), 191.0f);
        int lx = min(max((int)fx, 0), RES - 2);
        int ly = min(max((int)fy, 0), RES - 2);
        int lz = min(max((int)fz, 0), RES - 2);
        const float wbx = fx - (float)lx, wax = 1.0f - wbx;
        const float wby = fy - (float)ly, way = 1.0f - wby;
        const float wbz = fz - (float)lz, waz = 1.0f - wbz;

        // Issue all 8 link loads up front (adjacent z pairs merge to b64).
        const uint32_t b00 = (uint32_t)((lx * RES + ly) * RES + lz);      // dx=0,dy=0
        const uint32_t b01 = b00 + RES;                                    // dx=0,dy=1
        const uint32_t b10 = b00 + RES * RES;                              // dx=1,dy=0
        const uint32_t b11 = b10 + RES;                                    // dx=1,dy=1
        const int l000 = links[b00],     l001 = links[b00 + 1];
        const int l010 = links[b01],     l011 = links[b01 + 1];
        const int l100 = links[b10],     l101 = links[b10 + 1];
        const int l110 = links[b11],     l111 = links[b11 + 1];

        const float wAA = wax * way, wAB = wax * wby;
        const float wBA = wbx * way, wBB = wbx * wby;

        // Branchless weights: zero when link invalid (reference: where(valid, data, 0)).
        const float w000 = (l000 >= 0) ? wAA * waz : 0.0f;
        const float w001 = (l001 >= 0) ? wAA * wbz : 0.0f;
        const float w010 = (l010 >= 0) ? wAB * waz : 0.0f;
        const float w011 = (l011 >= 0) ? wAB * wbz : 0.0f;
        const float w100 = (l100 >= 0) ? wBA * waz : 0.0f;
        const float w101 = (l101 >= 0) ? wBA * wbz : 0.0f;
        const float w110 = (l110 >= 0) ? wBB * waz : 0.0f;
        const float w111 = (l111 >= 0) ? wBB * wbz : 0.0f;

        const uint32_t i000 = (uint32_t)max(l000, 0), i001 = (uint32_t)max(l001, 0);
        const uint32_t i010 = (uint32_t)max(l010, 0), i011 = (uint32_t)max(l011, 0);
        const uint32_t i100 = (uint32_t)max(l100, 0), i101 = (uint32_t)max(l101, 0);
        const uint32_t i110 = (uint32_t)max(l110, 0), i111 = (uint32_t)max(l111, 0);

        // Unconditional gathers: no exec-mask branches, so the scheduler can
        // keep multiple 108B row gathers in flight (MLP >> serialized form).
        gather_acc(w000, i000, density, sh, sigma, acc);
        gather_acc(w001, i001, density, sh, sigma, acc);
        gather_acc(w010, i010, density, sh, sigma, acc);
        gather_acc(w011, i011, density, sh, sigma, acc);
        gather_acc(w100, i100, density, sh, sigma, acc);
        gather_acc(w101, i101, density, sh, sigma, acc);
        gather_acc(w110, i110, density, sh, sigma, acc);
        gather_acc(w111, i111, density, sh, sigma, acc);

        out_sigma[pt] = sigma;  // coalesced b32 per lane
    }

    // -------- rgb writeout --------
    const int waveBasePt = blockIdx.x * BLOCK + wv * WAVE;
    const bool fullWave = (waveBasePt + WAVE) <= npts;

    if (fullWave) {
        // Stage: lane L writes dwords [L*27, L*27+27). Stride 27 (odd) -> 32
        // distinct banks per step, conflict-free.
#pragma unroll
        for (int c = 0; c < SH_DIM; ++c) stage[wv][lane * SH_DIM + c] = acc[c];

        // DS writes and the async engine use different counters: drain DScnt
        // before the async engine reads LDS.
        asm volatile("s_wait_dscnt 0" ::: "memory");

        // Wave region = 216 float4 chunks -> 7 async b128 bursts of 512 B.
        const uint64_t gbase = (uint64_t)(uintptr_t)(out_rgb + (size_t)waveBasePt * SH_DIM);
        const uint32_t lbase = (uint32_t)(uintptr_t)(&stage[wv][0]);
#pragma unroll
        for (int j = 0; j < 7; ++j) {
            const int chunk = j * WAVE + lane;
            if (chunk < V4_PER_WV) {
                uint64_t g = gbase + (uint64_t)chunk * 16u;
                uint32_t l = lbase + (uint32_t)chunk * 16u;
                asm volatile("global_store_async_from_lds_b128 %0, %1, off"
                             :: "v"(g), "v"(l) : "memory");
            }
        }
        asm volatile("s_wait_asynccnt 0" ::: "memory");
    } else if (active) {
        // Partial tail wave: direct register stores.
        float* dst = out_rgb + (size_t)pt * SH_DIM;
#pragma unroll
        for (int c = 0; c < SH_DIM; ++c) dst[c] = acc[c];
    }
}

extern "C" void kernel_launch(void* const* d_in, const int* in_sizes, int n_in,
                              void* d_out, int out_size, void* d_ws, size_t ws_size,
                              hipStream_t stream) {
    const float* points  = (const float*)d_in[0];  // N*3
    const float* density = (const float*)d_in[1];  // CAP*1
    const float* sh      = (const float*)d_in[2];  // CAP*27
    const int*   links   = (const int*)d_in[3];    // 192^3

    const int npts = in_sizes[0] / 3;
    float* out_sigma = (float*)d_out;              // first N floats
    float* out_rgb   = (float*)d_out + npts;       // then N*27 floats

    const int blocks = (npts + BLOCK - 1) / BLOCK;
    trilerp_sh_kernel<<<blocks, BLOCK, 0, stream>>>(points, density, sh, links,
                                                    out_sigma, out_rgb, npts);
}